// WaveletSeasonalDecomposer_43181601194295
// MI455X (gfx1250) — compile-verified
//
#include <hip/hip_runtime.h>

// CDNA5 / gfx1250 wave32 WMMA types
typedef __attribute__((ext_vector_type(2))) float v2f;
typedef __attribute__((ext_vector_type(8))) float v8f;

#define S_LEN   4096
#define S_MASK  4095
#define ROWS    16          // batch rows per workgroup (= WMMA M)
#define CW      1024        // seasonal columns per workgroup
#define XW      1200        // x tile width actually used  [n0-48, n0+1151]
#define XSTR    1204        // x tile LDS row stride (conflict-free, 16B aligned)
#define NA3T    9           // 16-wide a3 output tiles per WG (144 coeffs)
#define A3STR   148         // a3 tile LDS row stride
#define K1STEPS 43          // step-1 K window = 172 = 43 x (K=4)
#define TAUT    64          // 16-wide seasonal output tiles per WG
#define BLOCK   256
#define NWAVES  8

// dec_lo = reverse(_REC_LO_DB4)
__device__ __constant__ float DEC_LO[8] = {
  -0.010597401784997278f, 0.032883011666982945f, 0.030841381835986965f,
  -0.18703481171888114f,  -0.02798376941698385f, 0.6308807679295904f,
   0.7148465705525415f,    0.23037781330885523f };

// LDS layout (float offsets)
#define OFF_X   0
#define OFF_A3  (ROWS*XSTR)                  // 19264
#define OFF_B1  (OFF_A3 + ROWS*A3STR)        // 21632 (8B aligned for v2f)
#define OFF_H   (OFF_B1 + K1STEPS*32*2)      // 24384
#define LDS_FLOATS (OFF_H + 64)              // 24448 floats = 97792 bytes

__global__ __launch_bounds__(BLOCK)
void wavelet_seasonal(const float* __restrict__ x,
                      float* __restrict__ seasonal,
                      float* __restrict__ residual)
{
  extern __shared__ float smem[];
  float* Xs  = smem + OFF_X;
  float* A3s = smem + OFF_A3;
  v2f*   B1t = reinterpret_cast<v2f*>(smem + OFF_B1);
  float* Hs  = smem + OFF_H;

  const int tid  = threadIdx.x;
  const int lane = tid & 31;
  const int wave = tid >> 5;
  const int mrow = lane & 15;   // A: M row / B,C,D: N column for this lane
  const int hih  = lane >> 4;   // half-wave selector
  const int cb   = hih * 2;     // K sub-offset {0,1} vs {2,3} per ISA layout

  const int n0   = blockIdx.x * CW;
  const int row0 = blockIdx.y * ROWS;

  // ---- stage x tile [n0-48, n0+1151] (periodic mod 4096) into LDS ----
  {
    const int bcol = n0 - 48 + S_LEN;
    for (int idx = tid; idx < ROWS * (XW / 4); idx += BLOCK) {
      const int r = idx / (XW / 4);
      const int c = idx - r * (XW / 4);
      const int col = (bcol + 4 * c) & S_MASK;         // never splits a float4
      float4 v = *reinterpret_cast<const float4*>(x + (size_t)(row0 + r) * S_LEN + col);
      *reinterpret_cast<float4*>(Xs + r * XSTR + 4 * c) = v;
    }
  }

  // ---- composite 3-level analysis lowpass: H[u], u in [0,50) ----
  // a3[i] = sum_u H[u] * x[(8i+7-u) mod N],  H[u] = sum_{4j3+2j2+j1=u} g[j1]g[j2]g[j3]
  if (tid < 64) {
    float a = 0.f;
    if (tid < 50) {
      for (int j3 = 0; j3 < 8; ++j3)
        for (int j2 = 0; j2 < 8; ++j2) {
          const int j1 = tid - 4 * j3 - 2 * j2;
          if (j1 >= 0 && j1 < 8)
            a += DEC_LO[j1] * DEC_LO[j2] * DEC_LO[j3];
        }
    }
    Hs[tid] = a;
  }
  __syncthreads();

  // ---- B1 (172x16 banded, tile-invariant): B1[k,n] = H[8n+51-k] ----
  for (int e = tid; e < K1STEPS * 32; e += BLOCK) {
    const int s = e >> 5, L = e & 31;
    const int n = L & 15, c0 = (L >> 4) * 2;
    const int u0 = 8 * n + 51 - (4 * s + c0);
    v2f b;
    b.x = ((unsigned)u0       < 50u) ? Hs[u0]     : 0.f;
    b.y = ((unsigned)(u0 - 1) < 50u) ? Hs[u0 - 1] : 0.f;
    B1t[e] = b;
  }

  // ---- B2 fragments (8x16, tile-invariant): B2[k,n] = H[8k+7-n] ----
  v2f b2s[2];
  #pragma unroll
  for (int s = 0; s < 2; ++s) {
    const int kb = 4 * s + cb;
    const int u0 = 8 * kb + 7 - mrow;
    const int u1 = 8 * (kb + 1) + 7 - mrow;
    v2f b;
    b.x = ((unsigned)u0 < 50u) ? Hs[u0] : 0.f;
    b.y = ((unsigned)u1 < 50u) ? Hs[u1] : 0.f;
    b2s[s] = b;
  }
  __syncthreads();

  // ---- step 1: a3 tile = H (stride-8, 50-tap) applied to x, as WMMA chains ----
  // a3 local index i_loc = 16j + n; x local window starts at 128j + 4
  for (int j = wave; j < NA3T; j += NWAVES) {
    const int pb = 128 * j + 4 + cb;
    v8f acc = {0.f, 0.f, 0.f, 0.f, 0.f, 0.f, 0.f, 0.f};
    for (int s = 0; s < K1STEPS; ++s) {
      const float* xa = Xs + mrow * XSTR + pb + 4 * s;   // ds_load_b64, conflict-free
      v2f a = { xa[0], xa[1] };
      v2f b = B1t[s * 32 + lane];
      acc = __builtin_amdgcn_wmma_f32_16x16x4_f32(
              /*neg_a=*/false, a, /*neg_b=*/false, b,
              /*c_mod=*/(short)0, acc, /*reuse_a=*/false, /*reuse_b=*/false);
    }
    #pragma unroll
    for (int r = 0; r < 8; ++r)
      A3s[(r + 8 * hih) * A3STR + 16 * j + mrow] = acc[r];
  }
  __syncthreads();

  // ---- step 2: low = H^T a3 (K=8 -> 2 WMMAs/tile); emit both outputs ----
  for (int tau = wave; tau < TAUT; tau += NWAVES) {
    v8f acc = {0.f, 0.f, 0.f, 0.f, 0.f, 0.f, 0.f, 0.f};
    #pragma unroll
    for (int s = 0; s < 2; ++s) {
      const float* aa = A3s + mrow * A3STR + 2 * tau + 4 * s + cb;
      v2f a = { aa[0], aa[1] };
      acc = __builtin_amdgcn_wmma_f32_16x16x4_f32(
              false, a, false, b2s[s], (short)0, acc, false, false);
    }
    const int col = n0 + 16 * tau + mrow;   // this lane's output column
    const int xl  = 16 * tau + mrow + 48;   // same column in the LDS x tile
    #pragma unroll
    for (int r = 0; r < 8; ++r) {
      const int m = r + 8 * hih;            // batch row within tile (D layout)
      const float lowv = acc[r];
      const float xv   = Xs[m * XSTR + xl];
      const size_t gi  = (size_t)(row0 + m) * S_LEN + col;
      seasonal[gi] = xv - lowv;             // x - H^T H x
      residual[gi] = lowv;                  // H^T H x
    }
  }
}

extern "C" void kernel_launch(void* const* d_in, const int* in_sizes, int n_in,
                              void* d_out, int out_size, void* d_ws, size_t ws_size,
                              hipStream_t stream) {
  const float* x = (const float*)d_in[0];
  const int total = in_sizes[0];
  const int Brows = total / S_LEN;          // 2048
  float* seasonal = (float*)d_out;
  float* residual = seasonal + (size_t)Brows * S_LEN;
  dim3 grid(S_LEN / CW, Brows / ROWS);      // (4, 128)
  wavelet_seasonal<<<grid, BLOCK, LDS_FLOATS * sizeof(float), stream>>>(
      x, seasonal, residual);
}